// PointMatchingLoss_56538949485362
// MI455X (gfx1250) — compile-verified
//
#include <hip/hip_runtime.h>
#include <hip/hip_bf16.h>

#define BB 8
#define VV 8
#define HH 768
#define WW 1024
#define MM ((HH / 8) * (WW / 8)) // 12288
#define NN (BB * VV)             // 64

typedef __attribute__((ext_vector_type(2))) float v2f;
typedef __attribute__((ext_vector_type(8))) float v8f;

__device__ __forceinline__ void inv4(const float m[16], float o[16]) {
    float inv[16];
    inv[0]  =  m[5]*m[10]*m[15] - m[5]*m[11]*m[14] - m[9]*m[6]*m[15] + m[9]*m[7]*m[14] + m[13]*m[6]*m[11] - m[13]*m[7]*m[10];
    inv[4]  = -m[4]*m[10]*m[15] + m[4]*m[11]*m[14] + m[8]*m[6]*m[15] - m[8]*m[7]*m[14] - m[12]*m[6]*m[11] + m[12]*m[7]*m[10];
    inv[8]  =  m[4]*m[9]*m[15]  - m[4]*m[11]*m[13] - m[8]*m[5]*m[15] + m[8]*m[7]*m[13] + m[12]*m[5]*m[11] - m[12]*m[7]*m[9];
    inv[12] = -m[4]*m[9]*m[14]  + m[4]*m[10]*m[13] + m[8]*m[5]*m[14] - m[8]*m[6]*m[13] - m[12]*m[5]*m[10] + m[12]*m[6]*m[9];
    inv[1]  = -m[1]*m[10]*m[15] + m[1]*m[11]*m[14] + m[9]*m[2]*m[15] - m[9]*m[3]*m[14] - m[13]*m[2]*m[11] + m[13]*m[3]*m[10];
    inv[5]  =  m[0]*m[10]*m[15] - m[0]*m[11]*m[14] - m[8]*m[2]*m[15] + m[8]*m[3]*m[14] + m[12]*m[2]*m[11] - m[12]*m[3]*m[10];
    inv[9]  = -m[0]*m[9]*m[15]  + m[0]*m[11]*m[13] + m[8]*m[1]*m[15] - m[8]*m[3]*m[13] - m[12]*m[1]*m[11] + m[12]*m[3]*m[9];
    inv[13] =  m[0]*m[9]*m[14]  - m[0]*m[10]*m[13] - m[8]*m[1]*m[14] + m[8]*m[2]*m[13] + m[12]*m[1]*m[10] - m[12]*m[2]*m[9];
    inv[2]  =  m[1]*m[6]*m[15]  - m[1]*m[7]*m[14]  - m[5]*m[2]*m[15] + m[5]*m[3]*m[14] + m[13]*m[2]*m[7]  - m[13]*m[3]*m[6];
    inv[6]  = -m[0]*m[6]*m[15]  + m[0]*m[7]*m[14]  + m[4]*m[2]*m[15] - m[4]*m[3]*m[14] - m[12]*m[2]*m[7]  + m[12]*m[3]*m[6];
    inv[10] =  m[0]*m[5]*m[15]  - m[0]*m[7]*m[13]  - m[4]*m[1]*m[15] + m[4]*m[3]*m[13] + m[12]*m[1]*m[7]  - m[12]*m[3]*m[5];
    inv[14] = -m[0]*m[5]*m[14]  + m[0]*m[6]*m[13]  + m[4]*m[1]*m[14] - m[4]*m[2]*m[13] - m[12]*m[1]*m[6]  + m[12]*m[2]*m[5];
    inv[3]  = -m[1]*m[6]*m[11]  + m[1]*m[7]*m[10]  + m[5]*m[2]*m[11] - m[5]*m[3]*m[10] - m[9]*m[2]*m[7]   + m[9]*m[3]*m[6];
    inv[7]  =  m[0]*m[6]*m[11]  - m[0]*m[7]*m[10]  - m[4]*m[2]*m[11] + m[4]*m[3]*m[10] + m[8]*m[2]*m[7]   - m[8]*m[3]*m[6];
    inv[11] = -m[0]*m[5]*m[11]  + m[0]*m[7]*m[9]   + m[4]*m[1]*m[11] - m[4]*m[3]*m[9]  - m[8]*m[1]*m[7]   + m[8]*m[3]*m[5];
    inv[15] =  m[0]*m[5]*m[10]  - m[0]*m[6]*m[9]   - m[4]*m[1]*m[10] + m[4]*m[2]*m[9]  + m[8]*m[1]*m[6]   - m[8]*m[2]*m[5];
    float det = m[0]*inv[0] + m[1]*inv[4] + m[2]*inv[8] + m[3]*inv[12];
    det = 1.0f / det;
    for (int i = 0; i < 16; ++i) o[i] = inv[i] * det;
}

__device__ __forceinline__ void inv3(const float a[9], float o[9]) {
    float c0 = a[4]*a[8] - a[5]*a[7];
    float c1 = a[5]*a[6] - a[3]*a[8];
    float c2 = a[3]*a[7] - a[4]*a[6];
    float id = 1.0f / (a[0]*c0 + a[1]*c1 + a[2]*c2);
    o[0] = c0*id; o[1] = (a[2]*a[7]-a[1]*a[8])*id; o[2] = (a[1]*a[5]-a[2]*a[4])*id;
    o[3] = c1*id; o[4] = (a[0]*a[8]-a[2]*a[6])*id; o[5] = (a[2]*a[3]-a[0]*a[5])*id;
    o[6] = c2*id; o[7] = (a[1]*a[6]-a[0]*a[7])*id; o[8] = (a[0]*a[4]-a[1]*a[3])*id;
}

// Kernel 1: per-camera fused transform [M3 | T3] = [K2*R*K1inv | K2*t], 12 floats each.
__global__ void pm_setup_kernel(const float* __restrict__ k1, const float* __restrict__ k2,
                                const float* __restrict__ p1g, const float* __restrict__ p2g,
                                float* __restrict__ trans) {
    int n = threadIdx.x;
    if (n >= NN) return;
    // _vb permutation: flat n = v*B + b maps to stored camera index b*V + v
    int b = n % BB, v = n / BB;
    int cam = b * VV + v;

    const float sc[3] = {(float)WW, (float)HH, 1.0f};
    float K1[9], K2[9];
    for (int r = 0; r < 3; ++r)
        for (int c = 0; c < 3; ++c) {
            K1[r*3+c] = k1[cam*9 + r*3 + c] * sc[r];
            K2[r*3+c] = k2[cam*9 + r*3 + c] * sc[r];
        }
    float P1[16], P2[16], P2i[16], rel[16];
    for (int i = 0; i < 16; ++i) { P1[i] = p1g[cam*16 + i]; P2[i] = p2g[cam*16 + i]; }
    inv4(P2, P2i);
    for (int r = 0; r < 4; ++r)
        for (int c = 0; c < 4; ++c) {
            float s = 0.f;
            for (int k = 0; k < 4; ++k) s += P2i[r*4+k] * P1[k*4+c];
            rel[r*4+c] = s;
        }
    float K1i[9]; inv3(K1, K1i);
    float RK[9];
    for (int r = 0; r < 3; ++r)
        for (int c = 0; c < 3; ++c) {
            float s = 0.f;
            for (int k = 0; k < 3; ++k) s += rel[r*4+k] * K1i[k*3+c];
            RK[r*3+c] = s;
        }
    float M3[9], T3[3];
    for (int r = 0; r < 3; ++r) {
        float tacc = 0.f;
        for (int c = 0; c < 3; ++c) {
            float s = 0.f;
            for (int k = 0; k < 3; ++k) s += K2[r*3+k] * RK[k*3+c];
            M3[r*3+c] = s;
        }
        for (int k = 0; k < 3; ++k) tacc += K2[r*3+k] * rel[k*4+3];
        T3[r] = tacc;
    }
    for (int i = 0; i < 9; ++i) trans[n*12 + i] = M3[i];
    for (int r = 0; r < 3; ++r) trans[n*12 + 9 + r] = T3[r];
}

// Kernel 2: one block per camera n. Each wave projects 16 points per
// V_WMMA_F32_16X16X4_F32: D(16x16) = A(16x4) @ B(4x16) with
//   A rows 0..2 = [M3 | T3],  B col j = [x_j, y_j, 1, 1/dg_j].
// D row i (i<3), col j = proj_i(point j) / dg_j  -> ratios give x2p, y2p.
__global__ __launch_bounds__(256) void pm_main_kernel(
    const float* __restrict__ depths, const int* __restrict__ mxy1,
    const float* __restrict__ mxy2, const float* __restrict__ trans,
    float* __restrict__ partial) {
    const int n    = blockIdx.x;
    const int tid  = threadIdx.x;
    const int lane = tid & 31;
    const int wave = tid >> 5;
    const int mrow = lane & 15;
    const bool lo  = lane < 16;

    // Build A fragment (16x4 f32, 2 VGPRs/lane): lanes 0-15 hold K=0,1; lanes 16-31 hold K=2,3.
    const float* T = trans + n * 12;
    float a0 = 0.f, a1 = 0.f;
    if (mrow < 3) {
        if (lo) { a0 = T[mrow*3 + 0]; a1 = T[mrow*3 + 1]; }
        else    { a0 = T[mrow*3 + 2]; a1 = T[9 + mrow]; }
    }
    v2f afrag; afrag.x = a0; afrag.y = a1;

    const long img_base = (long)((n % BB) * VV + (n / BB)) * (long)(HH * WW);
    const int2*   m1 = (const int2*)mxy1 + (long)n * MM;
    const float2* m2 = (const float2*)mxy2 + (long)n * MM;

    float acc = 0.f;
    const int stride = (256 >> 5) * 16; // 8 waves * 16 pts = 128
    for (int base = wave * 16; base < MM; base += stride) {
        const int p = base + mrow;         // low & high half-lanes cover same 16 points
        const int2 xy = m1[p];
        float invd = 1.0f;
        if (!lo) {                         // only high lanes gather depth (feeds B row 3)
            float dg = depths[img_base + (long)xy.y * WW + xy.x];
            invd = 1.0f / dg;
        }
        v2f bfrag;                         // B rows: K=0 -> x, K=1 -> y (low lanes); K=2 -> 1, K=3 -> 1/dg (high lanes)
        bfrag.x = lo ? (float)xy.x : 1.0f;
        bfrag.y = lo ? (float)xy.y : invd;

        v8f c = {};
        c = __builtin_amdgcn_wmma_f32_16x16x4_f32(
            /*neg_a=*/false, afrag, /*neg_b=*/false, bfrag,
            /*c_mod=*/(short)0, c, /*reuse_a=*/false, /*reuse_b=*/false);

        if (lo) {                          // lane j (j<16): c[0..2] = proj of point j (scaled by 1/dg)
            const float2 gt = m2[p];
            const float rz = 1.0f / c[2];
            const float xp = c[0] * rz;
            const float yp = c[1] * rz;
            acc += fabsf(gt.x - xp) + fabsf(gt.y - yp);
        }
    }

    __shared__ float red[256];
    red[tid] = acc;
    __syncthreads();
    for (int s = 128; s > 0; s >>= 1) {    // fixed-order tree: deterministic
        if (tid < s) red[tid] += red[tid + s];
        __syncthreads();
    }
    if (tid == 0) partial[n] = red[0];
}

// Kernel 3: fixed-order reduce of the 64 per-camera partials, apply 1/(M*N).
__global__ void pm_reduce_kernel(const float* __restrict__ partial, float* __restrict__ out) {
    __shared__ float red[NN];
    int t = threadIdx.x;
    red[t] = partial[t];
    __syncthreads();
    for (int s = NN / 2; s > 0; s >>= 1) {
        if (t < s) red[t] += red[t + s];
        __syncthreads();
    }
    if (t == 0) out[0] = red[0] * (1.0f / ((float)MM * (float)NN));
}

extern "C" void kernel_launch(void* const* d_in, const int* in_sizes, int n_in,
                              void* d_out, int out_size, void* d_ws, size_t ws_size,
                              hipStream_t stream) {
    const float* depths = (const float*)d_in[0];
    const float* k1     = (const float*)d_in[1];
    const float* k2     = (const float*)d_in[2];
    const float* poses1 = (const float*)d_in[3];
    const float* poses2 = (const float*)d_in[4];
    const int*   mxy1   = (const int*)d_in[5];
    const float* mxy2   = (const float*)d_in[6];

    float* trans   = (float*)d_ws;        // NN*12 floats
    float* partial = trans + NN * 12;     // NN floats

    pm_setup_kernel<<<1, NN, 0, stream>>>(k1, k2, poses1, poses2, trans);
    pm_main_kernel<<<NN, 256, 0, stream>>>(depths, mxy1, mxy2, trans, partial);
    pm_reduce_kernel<<<1, NN, 0, stream>>>(partial, (float*)d_out);
}